// MeasureDistance_12781822673011
// MI455X (gfx1250) — compile-verified
//
#include <hip/hip_runtime.h>
#include <math.h>

typedef __attribute__((ext_vector_type(2))) float v2f;
typedef __attribute__((ext_vector_type(8))) float v8f;

#define BATCH 8
#define NPTS  2048
#define DIM   32
#define TILES (NPTS / 16)        // 128 column tiles
#define BN    (BATCH * NPTS)     // 16384

// ws layout (floats):
//  [0, BN)       : NX = 0.5*||x_i||^2
//  [BN, 2BN)     : NY = 0.5*||y_i||^2
//  [2BN, 10BN)   : F[set][which], set in {0,1}, which in {0:xx,1:yx,2:xy,3:yy}
//  [10BN, 14BN)  : FE[which] extrapolated potentials

__device__ __forceinline__ float* Fbuf(float* ws, int set, int which) {
    return ws + (size_t)2 * BN + (size_t)(set * 4 + which) * BN;
}

__global__ void sink_init(const float* __restrict__ x, const float* __restrict__ y,
                          float* __restrict__ ws) {
    int idx = blockIdx.x * blockDim.x + threadIdx.x;
    if (idx >= BN) return;
    const float* xr = x + (size_t)idx * DIM;
    const float* yr = y + (size_t)idx * DIM;
    float sx = 0.f, sy = 0.f;
    #pragma unroll
    for (int d = 0; d < DIM; ++d) { sx += xr[d] * xr[d]; sy += yr[d] * yr[d]; }
    ws[idx]      = 0.5f * sx;   // NX (pre-halved)
    ws[BN + idx] = 0.5f * sy;   // NY
    #pragma unroll
    for (int w = 0; w < 4; ++w) Fbuf(ws, 0, w)[idx] = 0.0f;   // zero initial potentials
}

// phase 0: Sinkhorn phase A  (blockIdx.y = 0:xx, 1:yx, 2:yy) -- update mode
// phase 1: Sinkhorn phase B  (xy, uses freshly written fyx)  -- update mode
// phase 2: extrapolation     (blockIdx.y = 0:fe_xx,1:fe_yx,2:fe_xy,3:fe_yy), out = -LSE
__global__ __launch_bounds__(256)
void sink_lse(const float* __restrict__ x, const float* __restrict__ y,
              const float* __restrict__ abias, const float* __restrict__ bbias,
              float* __restrict__ ws, int phase, int cur) {
    const int bat   = blockIdx.z;
    const int mi    = blockIdx.y;
    const int wave  = threadIdx.x >> 5;
    const int lane  = threadIdx.x & 31;
    const int half  = lane >> 4;     // 0 or 1
    const int lan15 = lane & 15;
    const int rowbase = blockIdx.x * 128 + wave * 16;
    const int nxt = 1 - cur;

    float* NX = ws;
    float* NY = ws + BN;
    float* FE = ws + (size_t)10 * BN;

    const float *rowp, *colp, *rn, *cn, *cb, *fcol, *fin;
    float* fout;
    int mode; // 0 = 0.5*fin - 0.5*LSE ; 1 = -LSE
    if (phase == 0) {
        mode = 0;
        if (mi == 0)      { rowp = x; colp = x; rn = NX; cn = NX; cb = abias;
                            fcol = Fbuf(ws, cur, 0); fin = Fbuf(ws, cur, 0); fout = Fbuf(ws, nxt, 0); }
        else if (mi == 1) { rowp = y; colp = x; rn = NY; cn = NX; cb = abias;
                            fcol = Fbuf(ws, cur, 2); fin = Fbuf(ws, cur, 1); fout = Fbuf(ws, nxt, 1); }
        else              { rowp = y; colp = y; rn = NY; cn = NY; cb = bbias;
                            fcol = Fbuf(ws, cur, 3); fin = Fbuf(ws, cur, 3); fout = Fbuf(ws, nxt, 3); }
    } else if (phase == 1) {
        mode = 0;
        rowp = x; colp = y; rn = NX; cn = NY; cb = bbias;
        fcol = Fbuf(ws, nxt, 1);   // NEW fyx from phase A of this iteration
        fin  = Fbuf(ws, cur, 2); fout = Fbuf(ws, nxt, 2);
    } else {
        mode = 1; fin = nullptr;
        if (mi == 0)      { rowp = x; colp = x; rn = NX; cn = NX; cb = abias;
                            fcol = Fbuf(ws, cur, 0); fout = FE + (size_t)0 * BN; }
        else if (mi == 1) { rowp = y; colp = x; rn = NY; cn = NX; cb = abias;
                            fcol = Fbuf(ws, cur, 2); fout = FE + (size_t)1 * BN; }
        else if (mi == 2) { rowp = x; colp = y; rn = NX; cn = NY; cb = bbias;
                            fcol = Fbuf(ws, cur, 1); fout = FE + (size_t)2 * BN; }
        else              { rowp = y; colp = y; rn = NY; cn = NY; cb = bbias;
                            fcol = Fbuf(ws, cur, 3); fout = FE + (size_t)3 * BN; }
    }
    // per-batch offsets
    rowp += (size_t)bat * NPTS * DIM;
    colp += (size_t)bat * NPTS * DIM;
    rn   += (size_t)bat * NPTS;  cn   += (size_t)bat * NPTS;
    cb   += (size_t)bat * NPTS;  fcol += (size_t)bat * NPTS;
    if (fin)  fin += (size_t)bat * NPTS;
    fout += (size_t)bat * NPTS;

    // ---- workgroup-wide Mc = max_j (bias_j + fcol_j) ------------------------
    __shared__ float sred[256];
    {
        float mloc = -1e30f;
        for (int i = threadIdx.x; i < NPTS; i += 256)
            mloc = fmaxf(mloc, cb[i] + fcol[i]);
        sred[threadIdx.x] = mloc;
        __syncthreads();
        for (int s2 = 128; s2 > 0; s2 >>= 1) {
            if (threadIdx.x < s2)
                sred[threadIdx.x] = fmaxf(sred[threadIdx.x], sred[threadIdx.x + s2]);
            __syncthreads();
        }
    }
    const float Mc = sred[0];

    // ---- preload A fragment: 16 rows x 32 dims ------------------------------
    // fp32 A(16x4) layout per k-step s: lane holds A[m=lane&15][k=4s+2*half+{0,1}]
    v2f afrag[8];
    const float* arow = rowp + (size_t)(rowbase + lan15) * DIM + 2 * half;
    #pragma unroll
    for (int s = 0; s < 8; ++s)
        afrag[s] = *(const v2f*)(arow + 4 * s);

    float hnx[8];   // 0.5*||x_row||^2 for the 8 rows this lane owns (M = r + 8*half)
    #pragma unroll
    for (int r = 0; r < 8; ++r) hnx[r] = rn[rowbase + r + 8 * half];

    float sum[8];
    #pragma unroll
    for (int r = 0; r < 8; ++r) sum[r] = 0.0f;

    // ---- double-buffered column-tile loop -----------------------------------
    v2f bf[2][8];
    float rcv[2], rfv[2], rnv[2];
    {   // issue tile 0
        const int mycol = lan15;
        const float* crow = colp + (size_t)mycol * DIM + 2 * half;
        #pragma unroll
        for (int s = 0; s < 8; ++s) bf[0][s] = *(const v2f*)(crow + 4 * s);
        rcv[0] = cb[mycol]; rfv[0] = fcol[mycol]; rnv[0] = cn[mycol];
    }

    #pragma unroll 2
    for (int t = 0; t < TILES; ++t) {
        const int cbuf = t & 1, nbuf = cbuf ^ 1;
        if (t + 1 < TILES) {   // issue loads for next tile before consuming this one
            const int mycol = (t + 1) * 16 + lan15;
            const float* crow = colp + (size_t)mycol * DIM + 2 * half;
            #pragma unroll
            for (int s = 0; s < 8; ++s) bf[nbuf][s] = *(const v2f*)(crow + 4 * s);
            rcv[nbuf] = cb[mycol]; rfv[nbuf] = fcol[mycol]; rnv[nbuf] = cn[mycol];
        }

        v8f acc = {0.f, 0.f, 0.f, 0.f, 0.f, 0.f, 0.f, 0.f};
        #pragma unroll
        for (int s = 0; s < 8; ++s)
            acc = __builtin_amdgcn_wmma_f32_16x16x4_f32(
                false, afrag[s], false, bf[cbuf][s], (short)0, acc, false, false);

        const float cfac = __expf(rcv[cbuf] + rfv[cbuf] - Mc);   // exp(colterm - Mc)
        const float hny  = rnv[cbuf];                            // 0.5*||y_col||^2
        #pragma unroll
        for (int r = 0; r < 8; ++r) {
            // kernel = exp(-max(0.5||x||^2 + 0.5||y||^2 - dot, 0))
            float t0 = fminf(acc[r] - (hnx[r] + hny), 0.0f);
            float kv = __expf(t0);
            sum[r] = fmaf(__expf(kv), cfac, sum[r]);             // exp(kv + colterm - Mc)
        }
    }

    // ---- combine partial sums across the 16 lanes of each half --------------
    #pragma unroll
    for (int r = 0; r < 8; ++r) {
        float s = sum[r];
        s += __shfl_xor(s, 1, 32);
        s += __shfl_xor(s, 2, 32);
        s += __shfl_xor(s, 4, 32);
        s += __shfl_xor(s, 8, 32);
        sum[r] = s;
    }

    if (lan15 == 0) {   // lanes 0 and 16 write rows 0..7 / 8..15 of the tile
        #pragma unroll
        for (int r = 0; r < 8; ++r) {
            int row = rowbase + r + 8 * half;
            float lse = Mc + __logf(sum[r]);
            fout[row] = (mode == 0) ? (0.5f * fin[row] - 0.5f * lse) : (-lse);
        }
    }
}

__global__ __launch_bounds__(256)
void sink_reduce(const float* __restrict__ abias, const float* __restrict__ bbias,
                 const float* __restrict__ ws, float* __restrict__ out) {
    const int bat = blockIdx.x;
    const float* FE   = ws + (size_t)10 * BN;
    const float* fexx = FE + (size_t)0 * BN + (size_t)bat * NPTS;
    const float* feyx = FE + (size_t)1 * BN + (size_t)bat * NPTS;
    const float* fexy = FE + (size_t)2 * BN + (size_t)bat * NPTS;
    const float* feyy = FE + (size_t)3 * BN + (size_t)bat * NPTS;
    const float* av = abias + (size_t)bat * NPTS;
    const float* bv = bbias + (size_t)bat * NPTS;

    __shared__ float red[256];
    float acc = 0.0f;
    for (int i = threadIdx.x; i < NPTS; i += 256) {
        acc += (fexy[i] - fexx[i]) * __expf(av[i]);
        acc += (feyx[i] - feyy[i]) * __expf(bv[i]);
    }
    red[threadIdx.x] = acc;
    __syncthreads();
    for (int s = 128; s > 0; s >>= 1) {
        if (threadIdx.x < s) red[threadIdx.x] += red[threadIdx.x + s];
        __syncthreads();
    }
    if (threadIdx.x == 0) out[bat] = red[0];   // EPSILON == 1
}

extern "C" void kernel_launch(void* const* d_in, const int* in_sizes, int n_in,
                              void* d_out, int out_size, void* d_ws, size_t ws_size,
                              hipStream_t stream) {
    (void)in_sizes; (void)n_in; (void)out_size; (void)ws_size;
    const float* x = (const float*)d_in[0];   // (8, 2048, 32)
    const float* a = (const float*)d_in[1];   // (8, 2048)
    const float* y = (const float*)d_in[2];   // (8, 2048, 32)
    const float* b = (const float*)d_in[3];   // (8, 2048)
    float* ws  = (float*)d_ws;
    float* out = (float*)d_out;               // (8,)

    sink_init<<<(BN + 255) / 256, 256, 0, stream>>>(x, y, ws);

    for (int it = 0; it < 10; ++it) {
        int cur = it & 1;
        sink_lse<<<dim3(NPTS / 128, 3, BATCH), 256, 0, stream>>>(x, y, a, b, ws, 0, cur);
        sink_lse<<<dim3(NPTS / 128, 1, BATCH), 256, 0, stream>>>(x, y, a, b, ws, 1, cur);
    }
    // after 10 iterations (even count) finals are back in set 0
    sink_lse<<<dim3(NPTS / 128, 4, BATCH), 256, 0, stream>>>(x, y, a, b, ws, 2, 0);
    sink_reduce<<<BATCH, 256, 0, stream>>>(a, b, ws, out);
}